// Transform_20693152432913
// MI455X (gfx1250) — compile-verified
//
#include <hip/hip_runtime.h>
#include <math.h>

// Problem constants (from the reference).
#define KN 50000    // nodes
#define KH 4        // heads
#define KC 32       // per-head channels
#define KD 128      // model dim (= H*C = in/out features)
#define KE 800000   // edges
#define ROWS_PER_BLOCK 80   // 5 x 16-row WMMA tiles; 50000 = 625 * 80 exact

typedef __attribute__((ext_vector_type(2))) float v2f;
typedef __attribute__((ext_vector_type(8))) float v8f;

// ---------------------------------------------------------------------------
// helpers
// ---------------------------------------------------------------------------
__device__ __forceinline__ void atomic_max_f32(float* addr, float val) {
  // Monotonic-order trick: int compare for >=0, uint compare for <0.
  if (val >= 0.0f) atomicMax((int*)addr, __float_as_int(val));
  else             atomicMin((unsigned int*)addr, __float_as_uint(val));
}

__device__ __forceinline__ void atomic_add_f32(float* addr, float val) {
  // Relaxed agent-scope fp32 add -> native global/flat_atomic_add_f32 on gfx1250.
  __hip_atomic_fetch_add(addr, val, __ATOMIC_RELAXED, __HIP_MEMORY_SCOPE_AGENT);
}

// ---------------------------------------------------------------------------
// amax = -inf, denom = 0
// ---------------------------------------------------------------------------
__global__ void init_softmax_state(float* __restrict__ amax,
                                   float* __restrict__ denom, int n) {
  int i = blockIdx.x * blockDim.x + threadIdx.x;
  if (i < n) {
    amax[i]  = -INFINITY;
    denom[i] = 0.0f;
  }
}

// ---------------------------------------------------------------------------
// Fused Q/K/V/Skip GEMM: OUT = X @ W + b for 4 weight matrices (blockIdx.y).
// 256 threads = 8 wave32; each wave owns a 16-column strip across 5 row tiles
// (80 rows) -> one staged W (64KB LDS) feeds 5 WMMA tiles per wave, and every
// B-operand load feeds 5 v_wmma_f32_16x16x4_f32 issues. The 5-way accumulator
// rotation keeps the matrix pipe full (each WMMA's C dep is 5 instrs back).
// ---------------------------------------------------------------------------
__global__ __launch_bounds__(256) void gemm_qkvs(
    const float* __restrict__ X,
    const float* __restrict__ W0, const float* __restrict__ B0,
    const float* __restrict__ W1, const float* __restrict__ B1,
    const float* __restrict__ W2, const float* __restrict__ B2,
    const float* __restrict__ W3, const float* __restrict__ B3,
    float* __restrict__ O0, float* __restrict__ O1,
    float* __restrict__ O2, float* __restrict__ O3) {
  __shared__ float sW[KD * KD];              // 64 KB
  __shared__ float sA[ROWS_PER_BLOCK * KD];  // 40 KB

  const float* Wsel[4] = {W0, W1, W2, W3};
  const float* Bsel[4] = {B0, B1, B2, B3};
  float*       Osel[4] = {O0, O1, O2, O3};
  const int mat = blockIdx.y;
  const float* __restrict__ W  = Wsel[mat];
  const float* __restrict__ Bi = Bsel[mat];
  float*       __restrict__ O  = Osel[mat];

  const int tid  = threadIdx.x;
  const int row0 = blockIdx.x * ROWS_PER_BLOCK;

  // Cooperative coalesced staging (b128 global loads -> b128 LDS stores).
  for (int i = tid; i < (KD * KD) / 4; i += 256)
    ((float4*)sW)[i] = ((const float4*)W)[i];
  for (int i = tid; i < (ROWS_PER_BLOCK * KD) / 4; i += 256)
    ((float4*)sA)[i] = ((const float4*)(X + (size_t)row0 * KD))[i];
  __syncthreads();

  const int wave = tid >> 5;
  const int lane = tid & 31;
  const int n    = lane & 15;       // B/C/D column within tile
  const int m    = lane & 15;       // A row within tile
  const int hi   = lane >> 4;       // lane-half selects K pair / row half
  const int col  = wave * 16 + n;   // global output column (8 waves * 16 = 128)

  v8f acc[5] = {};
#pragma unroll
  for (int kk = 0; kk < KD; kk += 4) {
    const int kA = kk + 2 * hi;
    v2f b;
    b.x = sW[kA * KD + col];        // B[kA,   col]
    b.y = sW[(kA + 1) * KD + col];  // B[kA+1, col]
#pragma unroll
    for (int rt = 0; rt < 5; ++rt) {
      v2f a = *(const v2f*)(sA + (rt * 16 + m) * KD + kA);  // A[rt*16+m, kA..]
      acc[rt] = __builtin_amdgcn_wmma_f32_16x16x4_f32(
          false, a, false, b, (short)0, acc[rt], false, false);
    }
  }

  const float bias = Bi[col];
#pragma unroll
  for (int rt = 0; rt < 5; ++rt) {
#pragma unroll
    for (int r = 0; r < 8; ++r) {
      const int row = row0 + rt * 16 + r + 8 * hi;  // C/D: vgpr r -> M=r+8*hi
      O[(size_t)row * KD + col] = acc[rt][r] + bias;
    }
  }
}

// ---------------------------------------------------------------------------
// Per (edge, head): alpha = <q[dst,h,:], k[src,h,:] + ea*We[h,:]> / sqrt(C)
// and segment-max into amax[dst,h].
// ---------------------------------------------------------------------------
__global__ void alpha_kernel(const float* __restrict__ q,
                             const float* __restrict__ k,
                             const int*   __restrict__ ei,
                             const float* __restrict__ ea,
                             const float* __restrict__ We,
                             float* __restrict__ alpha,
                             float* __restrict__ amax) {
  int i = blockIdx.x * blockDim.x + threadIdx.x;   // E*H = 3.2M, exact grid
  int e = i >> 2;
  int h = i & 3;
  int src = ei[e];
  int dst = ei[KE + e];
  float eav = ea[e];
  const float4* qv = (const float4*)(q + (size_t)dst * KD + h * KC);
  const float4* kv = (const float4*)(k + (size_t)src * KD + h * KC);
  const float4* wv = (const float4*)(We + h * KC);
  float s = 0.0f;
#pragma unroll
  for (int t = 0; t < 8; ++t) {
    float4 qq = qv[t], kk = kv[t], ww = wv[t];
    s += qq.x * (kk.x + eav * ww.x) + qq.y * (kk.y + eav * ww.y) +
         qq.z * (kk.z + eav * ww.z) + qq.w * (kk.w + eav * ww.w);
  }
  float al = s * 0.17677669529663687f;   // 1/sqrt(32)
  alpha[i] = al;
  atomic_max_f32(&amax[dst * KH + h], al);
}

// ---------------------------------------------------------------------------
// Per (edge, head): ex = exp(alpha - amax[dst]); alpha <- ex (in place);
// denom[dst,h] += ex.
// ---------------------------------------------------------------------------
__global__ void exp_kernel(const int* __restrict__ ei,
                           float* __restrict__ alpha,
                           const float* __restrict__ amax,
                           float* __restrict__ denom) {
  int i = blockIdx.x * blockDim.x + threadIdx.x;
  int e = i >> 2;
  int h = i & 3;
  int dst = ei[KE + e];
  float ex = __expf(alpha[i] - amax[dst * KH + h]);
  alpha[i] = ex;
  atomic_add_f32(&denom[dst * KH + h], ex);
}

// ---------------------------------------------------------------------------
// Per (edge, 4-channel chunk): msg = (v[src,c]+ea*We[c]) * ex/(denom+eps),
// scatter-added into the skip buffer (which already holds x@Ws + bs).
// ---------------------------------------------------------------------------
__global__ void msg_kernel(const int* __restrict__ ei,
                           const float* __restrict__ ea,
                           const float* __restrict__ v,
                           const float* __restrict__ We,
                           const float* __restrict__ exw,
                           const float* __restrict__ denom,
                           float* __restrict__ outacc) {
  int i = blockIdx.x * blockDim.x + threadIdx.x;   // E*32 = 25.6M, exact grid
  int e = i >> 5;
  int j = i & 31;
  int c = j * 4;          // channel chunk, stays inside one head (h = j>>3)
  int h = j >> 3;
  int src = ei[e];
  int dst = ei[KE + e];
  float w = exw[e * KH + h] / (denom[dst * KH + h] + 1e-16f);
  float eav = ea[e];
  float4 vv = *(const float4*)(v + (size_t)src * KD + c);
  float4 ww = *(const float4*)(We + c);
  float* o = outacc + (size_t)dst * KD + c;
  atomic_add_f32(o + 0, (vv.x + eav * ww.x) * w);
  atomic_add_f32(o + 1, (vv.y + eav * ww.y) * w);
  atomic_add_f32(o + 2, (vv.z + eav * ww.z) * w);
  atomic_add_f32(o + 3, (vv.w + eav * ww.w) * w);
}

// ---------------------------------------------------------------------------
// out = relu(in)  (both layers end with relu in the reference)
// ---------------------------------------------------------------------------
__global__ void relu_copy(const float* __restrict__ in,
                          float* __restrict__ out, int n) {
  int i = blockIdx.x * blockDim.x + threadIdx.x;
  if (i < n) out[i] = fmaxf(in[i], 0.0f);
}

// ---------------------------------------------------------------------------
// launch
// ---------------------------------------------------------------------------
extern "C" void kernel_launch(void* const* d_in, const int* in_sizes, int n_in,
                              void* d_out, int out_size, void* d_ws,
                              size_t ws_size, hipStream_t stream) {
  (void)in_sizes; (void)n_in; (void)out_size; (void)ws_size;

  const float* x  = (const float*)d_in[0];
  const int*   ei = (const int*)d_in[1];     // [2, E]
  const float* ea = (const float*)d_in[2];   // [E]

  const float* Wq1 = (const float*)d_in[3];  const float* bq1 = (const float*)d_in[4];
  const float* Wk1 = (const float*)d_in[5];  const float* bk1 = (const float*)d_in[6];
  const float* Wv1 = (const float*)d_in[7];  const float* bv1 = (const float*)d_in[8];
  const float* We1 = (const float*)d_in[9];
  const float* Ws1 = (const float*)d_in[10]; const float* bs1 = (const float*)d_in[11];

  const float* Wq2 = (const float*)d_in[12]; const float* bq2 = (const float*)d_in[13];
  const float* Wk2 = (const float*)d_in[14]; const float* bk2 = (const float*)d_in[15];
  const float* Wv2 = (const float*)d_in[16]; const float* bv2 = (const float*)d_in[17];
  const float* We2 = (const float*)d_in[18];
  const float* Ws2 = (const float*)d_in[19]; const float* bs2 = (const float*)d_in[20];

  float* out = (float*)d_out;

  // Workspace carve-up (floats). Total ~35.6M floats = ~143 MB.
  float* ws    = (float*)d_ws;
  float* q     = ws;
  float* k     = q     + (size_t)KN * KD;
  float* v     = k     + (size_t)KN * KD;
  float* sk    = v     + (size_t)KN * KD;   // skip output == message accumulator
  float* alpha = sk    + (size_t)KN * KD;   // [E,H]; reused in place for exp
  float* amax  = alpha + (size_t)KE * KH;
  float* denom = amax  + (size_t)KN * KH;
  float* hbuf  = denom + (size_t)KN * KH;   // layer-1 hidden activations

  auto run_layer = [&](const float* xin, const float* Wq, const float* bq,
                       const float* Wk, const float* bk, const float* Wv,
                       const float* bv, const float* We, const float* Wsk,
                       const float* bsk, float* outbuf) {
    init_softmax_state<<<(KN * KH + 255) / 256, 256, 0, stream>>>(amax, denom,
                                                                  KN * KH);
    gemm_qkvs<<<dim3(KN / ROWS_PER_BLOCK, 4), 256, 0, stream>>>(
        xin, Wq, bq, Wk, bk, Wv, bv, Wsk, bsk, q, k, v, sk);
    alpha_kernel<<<(KE * KH) / 256, 256, 0, stream>>>(q, k, ei, ea, We, alpha,
                                                      amax);
    exp_kernel<<<(KE * KH) / 256, 256, 0, stream>>>(ei, alpha, amax, denom);
    msg_kernel<<<(KE * 32) / 256, 256, 0, stream>>>(ei, ea, v, We, alpha,
                                                    denom, sk);
    relu_copy<<<(KN * KD) / 256, 256, 0, stream>>>(sk, outbuf, KN * KD);
  };

  run_layer(x,    Wq1, bq1, Wk1, bk1, Wv1, bv1, We1, Ws1, bs1, hbuf);
  run_layer(hbuf, Wq2, bq2, Wk2, bk2, Wv2, bv2, We2, Ws2, bs2, out);
}